// FP_53060025974802
// MI455X (gfx1250) — compile-verified
//
#include <hip/hip_runtime.h>

typedef float v2f __attribute__((ext_vector_type(2)));
typedef float v8f __attribute__((ext_vector_type(8)));
typedef float v4f __attribute__((ext_vector_type(4)));
typedef int   v4i __attribute__((ext_vector_type(4)));

#define BN_EPS 1e-5f

#if defined(__gfx1250__) && __has_builtin(__builtin_amdgcn_global_load_async_to_lds_b128) && __has_builtin(__builtin_amdgcn_s_wait_asynccnt)
#define USE_ASYNC 1
typedef __attribute__((address_space(1))) v4i gv4i;   // global int32x4
typedef __attribute__((address_space(3))) v4i lv4i;   // LDS int32x4
#else
#define USE_ASYNC 0
#endif

// -------------------------------------------------------------------------
// Kernel 1: 3-NN inverse-distance interpolation (b,n,3)x(b,m,3)->(b,n,256)
// One thread per small point; big coords + norms staged in LDS.
// -------------------------------------------------------------------------
__global__ __launch_bounds__(128) void interp_kernel(
    const float* __restrict__ co_small,   // (B, N, 3)
    const float* __restrict__ co_big,     // (B, M, 3)
    const float* __restrict__ fea_big,    // (B, M, C)
    float* __restrict__ interp,           // (B*N, C)
    int N, int M, int C)
{
  __shared__ float sx[1024], sy[1024], sz[1024], sn[1024];
  const int b   = blockIdx.y;
  const int tid = threadIdx.x;

  for (int i = tid; i < M; i += blockDim.x) {
    float x = co_big[((size_t)(b * M + i)) * 3 + 0];
    float y = co_big[((size_t)(b * M + i)) * 3 + 1];
    float z = co_big[((size_t)(b * M + i)) * 3 + 2];
    sx[i] = x; sy[i] = y; sz[i] = z;
    sn[i] = x * x + y * y + z * z;
  }
  __syncthreads();

  const int p  = blockIdx.x * blockDim.x + tid;   // point index within batch
  const size_t row = (size_t)b * N + p;
  const float px = co_small[row * 3 + 0];
  const float py = co_small[row * 3 + 1];
  const float pz = co_small[row * 3 + 2];
  const float ss = px * px + py * py + pz * pz;

  // top-3 smallest distances (expansion form, matching the reference's
  // |a|^2 + |b|^2 - 2 a.b arithmetic; earlier index wins ties via strict <)
  float d0 = 1e30f, d1 = 1e30f, d2 = 1e30f;
  int   i0 = 0,     i1 = 0,     i2 = 0;
  for (int m = 0; m < M; ++m) {
    float d = ss + sn[m] - 2.0f * (px * sx[m] + py * sy[m] + pz * sz[m]);
    if (d < d0)      { d2 = d1; i2 = i1; d1 = d0; i1 = i0; d0 = d; i0 = m; }
    else if (d < d1) { d2 = d1; i2 = i1; d1 = d;  i1 = m; }
    else if (d < d2) { d2 = d;  i2 = m; }
  }

  const bool  isbig = (d0 == 0.0f);
  const float w0 = 1.0f / d0, w1 = 1.0f / d1, w2 = 1.0f / d2;
  const float winv = 1.0f / (w0 + w1 + w2);

  const float* f0 = fea_big + ((size_t)b * M + i0) * C;
  const float* f1 = fea_big + ((size_t)b * M + i1) * C;
  const float* f2 = fea_big + ((size_t)b * M + i2) * C;
  float* o = interp + row * C;

  for (int c = 0; c < C; ++c) {
    float a = f0[c];
    float v = isbig ? a : (w0 * a + w1 * f1[c] + w2 * f2[c]) * winv;
    o[c] = v;
  }
}

// -------------------------------------------------------------------------
// GEMM helpers: async staging of a 64(N) x 32(K) tile of W into LDS.
// LDS row stride 36 floats: keeps 16B alignment for B128 writes and is
// bank-conflict-free (36*d % 64 distinct for d in [0,16)).
// -------------------------------------------------------------------------
#define KC        32
#define LDS_LDB   36

__device__ __forceinline__ void stage_chunk(
    const float* __restrict__ W, int K, int nBase, int k0,
    float* lbuf, int tid)
{
#pragma unroll
  for (int pass = 0; pass < 4; ++pass) {
    const int r    = pass * 16 + (tid >> 3);   // 0..63 (N row of the tile)
    const int col4 = (tid & 7) * 4;            // 0..28 (K within chunk)
    const float* g = W + (size_t)(nBase + r) * K + k0 + col4;
    float* l = lbuf + r * LDS_LDB + col4;
#if USE_ASYNC
    __builtin_amdgcn_global_load_async_to_lds_b128((gv4i*)g, (lv4i*)l, 0, 0);
#else
    *(v4f*)l = *(const v4f*)g;                 // sync fallback
#endif
  }
}

template <int N_>
__device__ __forceinline__ void wait_async_le() {
#if USE_ASYNC
  __builtin_amdgcn_s_wait_asynccnt(N_);
#endif
}

// -------------------------------------------------------------------------
// Fused GEMM + folded-BN + ReLU using V_WMMA_F32_16X16X4_F32, with the
// B (weight) tile double-buffered in LDS via async global->LDS DMA.
//   out[m, n] = relu( (sum_k A[m,k] * W[n,k]) * sc[n] + bi[n] )
// A columns [0,K0) come from A0 (ld=K0), columns [K0,K) from A1 (ld=K-K0).
// Block = 128 threads = 4 waves; wave w owns rows [mBase+16w, +16),
// each wave computes a 16x64 tile (4 accumulators across N).
// -------------------------------------------------------------------------
__global__ __launch_bounds__(128) void gemm_bn_relu_kernel(
    const float* __restrict__ A0, int K0,
    const float* __restrict__ A1,
    int K,
    const float* __restrict__ W,      // (256, K) row-major
    const float* __restrict__ bias,
    const float* __restrict__ gamma,
    const float* __restrict__ betap,
    const float* __restrict__ meanp,
    const float* __restrict__ varp,
    float* __restrict__ out)          // (Mtotal, 256)
{
  __shared__ float smem[2][64 * LDS_LDB];      // ~18 KB double buffer

  const int tid  = threadIdx.x;
  const int lane = tid & 31;
  const int wave = tid >> 5;           // 0..3
  const int hi   = lane >> 4;          // 0/1: K sub-pair on input, +8 rows on D
  const int l16  = lane & 15;

  const int  K1    = K - K0;
  const int  nBase = blockIdx.x * 64;
  const long mBase = (long)blockIdx.y * 64 + (long)wave * 16;

  const long  aRow = mBase + l16;
  const float* a0p = A0 ? (A0 + aRow * K0) : (const float*)nullptr;
  const float* a1p = A1 + aRow * K1;

  const int nChunks = K / KC;

  // prologue: stage chunk 0
  stage_chunk(W, K, nBase, 0, &smem[0][0], tid);

  v8f acc[4] = {};

  for (int c = 0; c < nChunks; ++c) {
    const int k0 = c * KC;
    const bool more = (c + 1 < nChunks);
    if (more) stage_chunk(W, K, nBase, k0 + KC, &smem[(c + 1) & 1][0], tid);

    // current chunk's 4 DMAs complete in-order; prefetched ones stay in flight
    if (more) wait_async_le<4>(); else wait_async_le<0>();
    __syncthreads();

    const float* bw = &smem[c & 1][0];
#pragma unroll
    for (int k = 0; k < KC; k += 4) {
      const int kk = k0 + k + 2 * hi;          // global K for this lane's A pair
      v2f a;
      if (kk < K0) a = *(const v2f*)(a0p + kk);
      else         a = *(const v2f*)(a1p + (kk - K0));
      if (k == 0) {
        // hint the A stream one chunk ahead
        if (kk + KC < K0)      __builtin_prefetch(a0p + kk + KC, 0, 1);
        else if (kk + KC < K)  __builtin_prefetch(a1p + (kk + KC - K0), 0, 1);
      }
      const int lk = k + 2 * hi;               // K within the LDS chunk
#pragma unroll
      for (int j = 0; j < 4; ++j) {
        v2f bv = *(const v2f*)(bw + (16 * j + l16) * LDS_LDB + lk);
        acc[j] = __builtin_amdgcn_wmma_f32_16x16x4_f32(
            /*neg_a=*/false, a, /*neg_b=*/false, bv,
            /*c_mod=*/(short)0, acc[j], /*reuse_a=*/false, /*reuse_b=*/false);
      }
    }
    __syncthreads();   // protect buffer c&1 before it is re-staged
  }

  // Epilogue: folded BN (scale/shift per output channel) + ReLU.
#pragma unroll
  for (int j = 0; j < 4; ++j) {
    const int ncol = nBase + 16 * j + l16;
    const float sc = gamma[ncol] * rsqrtf(varp[ncol] + BN_EPS);
    const float bi = (bias[ncol] - meanp[ncol]) * sc + betap[ncol];
#pragma unroll
    for (int r = 0; r < 8; ++r) {
      const long mRow = mBase + r + 8 * hi;    // C/D layout: VGPR r, halves
      float v = acc[j][r] * sc + bi;
      out[mRow * 256 + ncol] = v > 0.0f ? v : 0.0f;
    }
  }
}

extern "C" void kernel_launch(void* const* d_in, const int* in_sizes, int n_in,
                              void* d_out, int out_size, void* d_ws, size_t ws_size,
                              hipStream_t stream) {
  const float* co_s  = (const float*)d_in[0];   // (16,4096,3)
  const float* fea_s = (const float*)d_in[1];   // (16,4096,128)
  const float* co_b  = (const float*)d_in[2];   // (16,1024,3)
  const float* fea_b = (const float*)d_in[3];   // (16,1024,256)
  const float* W0    = (const float*)d_in[4];   // (256,384)
  const float* b0    = (const float*)d_in[5];
  const float* g0    = (const float*)d_in[6];
  const float* be0   = (const float*)d_in[7];
  const float* m0    = (const float*)d_in[8];
  const float* v0    = (const float*)d_in[9];
  const float* W1    = (const float*)d_in[10];  // (256,256)
  const float* b1    = (const float*)d_in[11];
  const float* g1    = (const float*)d_in[12];
  const float* be1   = (const float*)d_in[13];
  const float* m1    = (const float*)d_in[14];
  const float* v1    = (const float*)d_in[15];

  const int B = 16, N = 4096, M = 1024, Cs = 128, Cb = 256, H = 256;

  float* interp = (float*)d_ws;                       // (B*N, 256)  = 64 MB
  float* y0     = interp + (size_t)B * N * Cb;        // (B*N, 256)  = 64 MB

  // 3-NN interpolation
  dim3 gi(N / 128, B);
  interp_kernel<<<gi, 128, 0, stream>>>(co_s, co_b, fea_b, interp, N, M, Cb);

  // Layer 0: A = [fea_small | interp], K = 384
  dim3 gg(256 / 64, (B * N) / 64);
  gemm_bn_relu_kernel<<<gg, 128, 0, stream>>>(fea_s, Cs, interp, Cs + Cb,
                                              W0, b0, g0, be0, m0, v0, y0);

  // Layer 1: A = y0, K = 256
  gemm_bn_relu_kernel<<<gg, 128, 0, stream>>>(nullptr, 0, y0, H,
                                              W1, b1, g1, be1, m1, v1,
                                              (float*)d_out);
}